// FeatureICA_60567628808847
// MI455X (gfx1250) — compile-verified
//
#include <hip/hip_runtime.h>

#define ICA_G 4
#define ICA_D 256
#define ICA_NITER 10
#define ICA_MASK_FLOOR 1e-5f
#define ICA_EPS 1e-5f
#define ROWP 260   // padded LDS row stride (floats): 16B-aligned rows, bank-shifted per group

typedef float v2f __attribute__((ext_vector_type(2)));
typedef float v4f __attribute__((ext_vector_type(4)));
typedef float v8f __attribute__((ext_vector_type(8)));

__device__ __forceinline__ float det3f(float a, float b, float c,
                                       float d, float e, float f,
                                       float g, float h, float i) {
  return a * (e * i - f * h) - b * (d * i - f * g) + c * (d * h - e * g);
}

// 3x3 minor of 4x4 M deleting row r, col c (r,c constant after unrolling)
__device__ __forceinline__ float minor3(const float M[4][4], int r, int c) {
  int rr[3], cc[3];
  int t = 0;
#pragma unroll
  for (int i = 0; i < 4; ++i) if (i != r) rr[t++] = i;
  t = 0;
#pragma unroll
  for (int j = 0; j < 4; ++j) if (j != c) cc[t++] = j;
  return det3f(M[rr[0]][cc[0]], M[rr[0]][cc[1]], M[rr[0]][cc[2]],
               M[rr[1]][cc[0]], M[rr[1]][cc[1]], M[rr[1]][cc[2]],
               M[rr[2]][cc[0]], M[rr[2]][cc[1]], M[rr[2]][cc[2]]);
}

__device__ __forceinline__ v8f wmma_f32_k4(v2f a, v2f b, v8f c) {
#if __has_builtin(__builtin_amdgcn_wmma_f32_16x16x4_f32)
  return __builtin_amdgcn_wmma_f32_16x16x4_f32(false, a, false, b,
                                               (short)0, c, false, false);
#else
  asm("v_wmma_f32_16x16x4_f32 %0, %1, %2, %0" : "+v"(c) : "v"(a), "v"(b));
  return c;
#endif
}

__global__ __launch_bounds__(32) void ica_wave_kernel(const float* __restrict__ x,
                                                      float* __restrict__ out,
                                                      int nprob) {
  __shared__ float Xs[ICA_G * ROWP];  // xg, one padded row per group
  __shared__ float Us[ICA_G * ROWP];  // contrast weights u, same layout
  __shared__ float Vs[16 * 4];        // all four V_k: Vs[(4k+g)*4 + h]

  const int lane = threadIdx.x;  // wave32: one wave per workgroup, one problem per block
  const int p = blockIdx.x;
  if (p >= nprob) return;
  const long long base = (long long)p * (ICA_G * ICA_D);

  // ---- stage xg into LDS (float4, fully coalesced) ----
#pragma unroll
  for (int i = 0; i < 8; ++i) {
    int flat = i * 128 + lane * 4;  // 0..1023
    int g = flat >> 8;
    int d = flat & 255;
    v4f v = *(const v4f*)(x + base + flat);
    *(v4f*)(&Xs[g * ROWP + d]) = v;
  }
  __syncthreads();

  // demixing matrix W, replicated in every lane (wave-uniform -> compiler scalarizes)
  float W[4][4];
#pragma unroll
  for (int i = 0; i < 4; ++i)
#pragma unroll
    for (int j = 0; j < 4; ++j) W[i][j] = (i == j) ? 1.0f : 0.0f;

  // q from the sigmoid re-parameterization (Q_INIT = 1.0); qexp = q - 2
  const float inv_q = -logf((2.0f - ICA_EPS) / (1.0f - ICA_EPS) - 1.0f);
  const float qv = (2.0f - ICA_EPS) / (1.0f + expf(-inv_q)) + ICA_EPS;
  const float qexp = qv - 2.0f;

  // WMMA fragment roles for this lane (f32 16x16x4 layout):
  // A row M = 4*k + g  ->  A[M][K] = u_k[d] * x_g[d]
  // B col N, h = N & 3 ->  B[K][N] = x_h[d]   (note: h == g == lane&3 per lane,
  //                                            so one x-row load feeds both A and B)
  // lanes 0-15 carry K=0,1 (d0, d0+1); lanes 16-31 carry K=2,3 (d0+2, d0+3)
  const int Mrow = lane & 15;
  const int kA = Mrow >> 2;
  const int gA = Mrow & 3;
  const int dOff = (lane < 16) ? 0 : 2;

#pragma unroll 1
  for (int it = 0; it < ICA_NITER; ++it) {
    // ---- phase A: y = W x ; u = max((|y|+eps)^qexp, floor) -> LDS ----
#pragma unroll
    for (int i = 0; i < 8; ++i) {
      int d = i * 32 + lane;
      float x0 = Xs[0 * ROWP + d], x1 = Xs[1 * ROWP + d];
      float x2 = Xs[2 * ROWP + d], x3 = Xs[3 * ROWP + d];
#pragma unroll
      for (int g = 0; g < 4; ++g) {
        float y = W[g][0] * x0 + W[g][1] * x1 + W[g][2] * x2 + W[g][3] * x3;
        float r = fabsf(y) + ICA_EPS;
        float u = exp2f(qexp * log2f(r));
        Us[g * ROWP + d] = fmaxf(u, ICA_MASK_FLOOR);
      }
    }
    __syncthreads();

    // ---- phase B: all four weighted covariances via WMMA f32 16x16x4 ----
    // Two independent accumulator chains (even/odd K-chunks) halve the serial
    // D->C dependency depth; summed elementwise afterwards (linearity).
    v8f acc0 = {};
    v8f acc1 = {};
#pragma unroll 8
    for (int c = 0; c < 64; c += 2) {
      {
        int d0 = 4 * c + dOff;
        v2f u2 = *(const v2f*)(&Us[kA * ROWP + d0]);
        v2f x2 = *(const v2f*)(&Xs[gA * ROWP + d0]);
        v2f a;
        a.x = u2.x * x2.x;
        a.y = u2.y * x2.y;
        acc0 = wmma_f32_k4(a, x2, acc0);
      }
      {
        int d1 = 4 * (c + 1) + dOff;
        v2f u2 = *(const v2f*)(&Us[kA * ROWP + d1]);
        v2f x2 = *(const v2f*)(&Xs[gA * ROWP + d1]);
        v2f a;
        a.x = u2.x * x2.x;
        a.y = u2.y * x2.y;
        acc1 = wmma_f32_k4(a, x2, acc1);
      }
    }
    // scatter D-fragment (scaled by 1/D) to Vs; only columns N<4 are needed
    if ((lane & 15) < 4) {
      const int h = lane & 3;
      const int mb = (lane < 16) ? 0 : 8;  // C/D layout: VGPR j -> M=j (lo) / j+8 (hi)
#pragma unroll
      for (int j = 0; j < 8; ++j)
        Vs[(mb + j) * 4 + h] = (acc0[j] + acc1[j]) * (1.0f / ICA_D);
    }
    __syncthreads();

    // ---- phase C: sequential per-k row update (wave-uniform -> SALU) ----
#pragma unroll
    for (int k = 0; k < 4; ++k) {
      float Vk[4][4];
#pragma unroll
      for (int i2 = 0; i2 < 4; ++i2)
#pragma unroll
        for (int j2 = 0; j2 < 4; ++j2) Vk[i2][j2] = Vs[(4 * k + i2) * 4 + j2];

      float WV[4][4];
#pragma unroll
      for (int i2 = 0; i2 < 4; ++i2)
#pragma unroll
        for (int j2 = 0; j2 < 4; ++j2)
          WV[i2][j2] = W[i2][0] * Vk[0][j2] + W[i2][1] * Vk[1][j2] +
                       W[i2][2] * Vk[2][j2] + W[i2][3] * Vk[3][j2];

      // w = solve(WV, e_k) = column k of inv(WV) via cofactors of row k
      float C[4];
#pragma unroll
      for (int j2 = 0; j2 < 4; ++j2) {
        float m = minor3(WV, k, j2);
        C[j2] = (((k + j2) & 1) ? -m : m);
      }
      float det = WV[k][0] * C[0] + WV[k][1] * C[1] + WV[k][2] * C[2] + WV[k][3] * C[3];
      float idet = 1.0f / det;
      float w[4];
#pragma unroll
      for (int j2 = 0; j2 < 4; ++j2) w[j2] = C[j2] * idet;
      // normalize: w /= sqrt(w^T V w) + eps
      float qf = 0.0f;
#pragma unroll
      for (int i2 = 0; i2 < 4; ++i2)
#pragma unroll
        for (int j2 = 0; j2 < 4; ++j2) qf += w[i2] * Vk[i2][j2] * w[j2];
      float dn = sqrtf(qf) + ICA_EPS;
#pragma unroll
      for (int j2 = 0; j2 < 4; ++j2) W[k][j2] = w[j2] / dn;
    }
    __syncthreads();
  }

  // ---- A = inv(W); scale_k = A[k][k] = minor_kk(W)/det(W) ----
  float detW = 0.0f;
#pragma unroll
  for (int j = 0; j < 4; ++j) {
    float m = minor3(W, 0, j);
    detW += W[0][j] * ((j & 1) ? -m : m);
  }
  float idetW = 1.0f / detW;
  float scale[4];
#pragma unroll
  for (int k = 0; k < 4; ++k) scale[k] = minor3(W, k, k) * idetW;

  // ---- y = W x, minimal-distortion projection, coalesced store ----
#pragma unroll
  for (int i = 0; i < 8; ++i) {
    int d = i * 32 + lane;
    float x0 = Xs[0 * ROWP + d], x1 = Xs[1 * ROWP + d];
    float x2 = Xs[2 * ROWP + d], x3 = Xs[3 * ROWP + d];
#pragma unroll
    for (int g = 0; g < 4; ++g) {
      float y = W[g][0] * x0 + W[g][1] * x1 + W[g][2] * x2 + W[g][3] * x3;
      out[base + g * ICA_D + d] = y * scale[g];
    }
  }
}

extern "C" void kernel_launch(void* const* d_in, const int* in_sizes, int n_in,
                              void* d_out, int out_size, void* d_ws, size_t ws_size,
                              hipStream_t stream) {
  (void)n_in; (void)out_size; (void)d_ws; (void)ws_size;
  const float* x = (const float*)d_in[0];
  float* out = (float*)d_out;
  const int nprob = in_sizes[0] / (ICA_G * ICA_D);  // B*T = 8192
  ica_wave_kernel<<<dim3(nprob), dim3(32), 0, stream>>>(x, out, nprob);
}